// ManoLayer_72353019068803
// MI455X (gfx1250) — compile-verified
//
#include <hip/hip_runtime.h>
#include <hip/hip_bf16.h>

// MANO forward for MI455X (gfx1250, wave32).
//  K0: build fused (K=148)x(2352) operand  [betas | pose_map | const] pre-swizzled
//      into WMMA K-pair-interleaved layout (one 8B load per operand per WMMA).
//  K1: pre-contract J_regressor into shapedirs/template (16x3x11 table).
//  K2: per-batch rig: pose PCA, Rodrigues, kinematic chain, results2, joint jtr.
//  K3: fused GEMM (v_wmma_f32_16x16x4_f32) + LBS skinning via LDS; writes verts
//      and the 5 fingertip jtr entries.

#define B_N    4096
#define V_N    778
#define J_N    16
#define NB_N   10
#define KTOT   148      // 10 betas + 135 pose_map + 1 const + 2 zero pad
#define KPAIRS 74
#define BMCOLS 2352     // 778*3 = 2334 padded up so tile 768..783 stays in-bounds
#define VPB    128      // vertices per block (8 waves x 16 vertices)
#define RES_STR 196     // LDS stride pads (bank de-conflict, 16 batches)
#define VP_STR  389

typedef float v2f __attribute__((ext_vector_type(2)));
typedef float v8f __attribute__((ext_vector_type(8)));

#if defined(__HIP_DEVICE_COMPILE__) && defined(__gfx1250__) && \
    __has_builtin(__builtin_amdgcn_wmma_f32_16x16x4_f32)
#define HAVE_WMMA 1
#else
#define HAVE_WMMA 0
#endif

__constant__ int c_parents[16] = {0,0,1,2,0,4,5,0,7,8,0,10,11,0,13,14};
// output slot in jtr for joint i (inverse of REORDER for values < 16)
__constant__ int c_jslot[16]   = {0,5,6,7,9,10,11,17,18,19,13,14,15,1,2,3};

// ---------------------------------------------------------------- K0: operand
__global__ void build_bmat_kernel(const float* __restrict__ sd,
                                  const float* __restrict__ pd,
                                  const float* __restrict__ vt,
                                  float* __restrict__ bmat)
{
  int idx = blockIdx.x * 256 + threadIdx.x;       // over KTOT*BMCOLS
  if (idx >= KTOT * BMCOLS) return;
  int k = idx / BMCOLS, c = idx - k * BMCOLS;
  float v = 0.f;
  if (c < V_N * 3) {
    if (k < NB_N)            v = sd[c * NB_N + k];          // shapedirs[v,d,k]
    else if (k < NB_N + 135) v = pd[c * 135 + (k - NB_N)];  // posedirs[v,d,p]
    else if (k == 145)       v = vt[c];                     // v_template
  }
  // K-pair interleave: element (k,c) -> ((k/2)*BMCOLS + c)*2 + (k&1)
  bmat[((size_t)(k >> 1) * BMCOLS + c) * 2 + (k & 1)] = v;
}

// --------------------------------------------- K1: JR pre-contraction (16x3x11)
__global__ void build_jsd_kernel(const float* __restrict__ jr,
                                 const float* __restrict__ sd,
                                 const float* __restrict__ vt,
                                 float* __restrict__ jsd)
{
  int t = blockIdx.x * 64 + threadIdx.x;
  if (t >= J_N * 33) return;
  int jt = t / 33, rem = t - jt * 33;
  int d = rem / 11, k = rem - d * 11;               // k==10 -> template term
  float s = 0.f;
  for (int v = 0; v < V_N; ++v) {
    float w = jr[jt * V_N + v];
    float x = (k < NB_N) ? sd[(v * 3 + d) * NB_N + k] : vt[v * 3 + d];
    s += w * x;
  }
  jsd[t] = s;
}

// ------------------------------------------------------------------ Rodrigues
__device__ __forceinline__ void rodrigues(float ax, float ay, float az, float R[9])
{
  float s0 = ax + 1e-8f, s1 = ay + 1e-8f, s2 = az + 1e-8f;
  float angle = sqrtf(s0 * s0 + s1 * s1 + s2 * s2);
  float inv = 1.0f / angle;
  float half = 0.5f * angle;
  float w = cosf(half);
  float sh = sinf(half);
  float x = ax * inv * sh, y = ay * inv * sh, z = az * inv * sh;
  float n = rsqrtf(w * w + x * x + y * y + z * z);
  w *= n; x *= n; y *= n; z *= n;
  float w2 = w * w, x2 = x * x, y2 = y * y, z2 = z * z;
  float wx = w * x, wy = w * y, wz = w * z;
  float xy = x * y, xz = x * z, yz = y * z;
  R[0] = w2 + x2 - y2 - z2; R[1] = 2.f * (xy - wz);     R[2] = 2.f * (wy + xz);
  R[3] = 2.f * (wz + xy);   R[4] = w2 - x2 + y2 - z2;   R[5] = 2.f * (yz - wx);
  R[6] = 2.f * (xz - wy);   R[7] = 2.f * (wx + yz);     R[8] = w2 - x2 - y2 + z2;
}

// --------------------------------------------------------------- K2: rig math
__global__ __launch_bounds__(64) void rig_kernel(
    const float* __restrict__ pc_all, const float* __restrict__ betas_all,
    const float* __restrict__ trans_all, const float* __restrict__ sc,
    const float* __restrict__ hm, const float* __restrict__ jsd,
    float* __restrict__ pose_ext, float* __restrict__ res2,
    float* __restrict__ jtr_out)
{
  __shared__ float Tl[64 * 192];                 // 16 transforms (3x4) / batch
  int b = blockIdx.x * 64 + threadIdx.x;
  float* T = &Tl[threadIdx.x * 192];
  const float* pc = pc_all + b * 48;

  float beta[10];
  for (int k = 0; k < 10; ++k) beta[k] = betas_all[b * 10 + k];

  // joints via pre-contracted regressor
  float jx[16], jy[16], jz[16];
  for (int i = 0; i < 16; ++i) {
    const float* a = jsd + i * 33;
    float sx = a[10], sy = a[11 + 10], sz = a[22 + 10];
    for (int k = 0; k < 10; ++k) {
      sx += a[k] * beta[k];
      sy += a[11 + k] * beta[k];
      sz += a[22 + k] * beta[k];
    }
    jx[i] = sx; jy[i] = sy; jz[i] = sz;
  }

  float* pe = pose_ext + (size_t)b * KTOT;       // fused-GEMM A row
  for (int k = 0; k < 10; ++k) pe[k] = beta[k];
  pe[145] = 1.f; pe[146] = 0.f; pe[147] = 0.f;

  float pcl[45];
  for (int c = 0; c < 45; ++c) pcl[c] = pc[3 + c];

  float R[9];
  rodrigues(pc[0], pc[1], pc[2], R);             // root
  T[0] = R[0]; T[1] = R[1]; T[2]  = R[2]; T[3]  = jx[0];
  T[4] = R[3]; T[5] = R[4]; T[6]  = R[5]; T[7]  = jy[0];
  T[8] = R[6]; T[9] = R[7]; T[10] = R[8]; T[11] = jz[0];

  for (int i = 1; i < 16; ++i) {
    int col = 3 * (i - 1);
    float px = hm[col + 0], py = hm[col + 1], pz = hm[col + 2];
    for (int c = 0; c < 45; ++c) {               // pose PCA (45x45)
      float v = pcl[c];
      px += v * sc[c * 45 + col + 0];
      py += v * sc[c * 45 + col + 1];
      pz += v * sc[c * 45 + col + 2];
    }
    rodrigues(px, py, pz, R);
    float* pm = pe + 10 + 9 * (i - 1);           // pose_map = R - I
    pm[0] = R[0] - 1.f; pm[1] = R[1];       pm[2] = R[2];
    pm[3] = R[3];       pm[4] = R[4] - 1.f; pm[5] = R[5];
    pm[6] = R[6];       pm[7] = R[7];       pm[8] = R[8] - 1.f;

    int p = c_parents[i];
    const float* Tp = T + p * 12;
    float lx = jx[i] - jx[p], ly = jy[i] - jy[p], lz = jz[i] - jz[p];
    float* Ti = T + i * 12;
    for (int r = 0; r < 3; ++r) {
      float a0 = Tp[r*4+0], a1 = Tp[r*4+1], a2 = Tp[r*4+2], a3 = Tp[r*4+3];
      Ti[r*4+0] = a0 * R[0] + a1 * R[3] + a2 * R[6];
      Ti[r*4+1] = a0 * R[1] + a1 * R[4] + a2 * R[7];
      Ti[r*4+2] = a0 * R[2] + a1 * R[5] + a2 * R[8];
      Ti[r*4+3] = a0 * lx + a1 * ly + a2 * lz + a3;
    }
  }

  float tr0 = trans_all[b*3+0], tr1 = trans_all[b*3+1], tr2 = trans_all[b*3+2];
  float* rb = res2 + (size_t)b * 192;
  for (int i = 0; i < 16; ++i) {
    const float* Ti = T + i * 12;
    float* jo = jtr_out + ((size_t)b * 21 + c_jslot[i]) * 3;   // joints of jtr
    jo[0] = (Ti[3]  + tr0) * 1000.f;
    jo[1] = (Ti[7]  + tr1) * 1000.f;
    jo[2] = (Ti[11] + tr2) * 1000.f;
    float cx = Ti[0]*jx[i] + Ti[1]*jy[i] + Ti[2]*jz[i];        // corr = R@j
    float cy = Ti[4]*jx[i] + Ti[5]*jy[i] + Ti[6]*jz[i];
    float cz = Ti[8]*jx[i] + Ti[9]*jy[i] + Ti[10]*jz[i];
    float* ro = rb + i * 12;
    ro[0]=Ti[0]; ro[1]=Ti[1]; ro[2]=Ti[2];  ro[3]=Ti[3]-cx;
    ro[4]=Ti[4]; ro[5]=Ti[5]; ro[6]=Ti[6];  ro[7]=Ti[7]-cy;
    ro[8]=Ti[8]; ro[9]=Ti[9]; ro[10]=Ti[10];ro[11]=Ti[11]-cz;
  }
}

// -------------------------------------------- K3: WMMA GEMM + fused skinning
__global__ __launch_bounds__(256) void skin_kernel(
    const float* __restrict__ bmat, const float* __restrict__ pose_ext,
    const float* __restrict__ res2, const float* __restrict__ trans_all,
    const float* __restrict__ wts, float* __restrict__ verts_out,
    float* __restrict__ jtr_out)
{
  __shared__ float A_lds[KPAIRS * 32];       // pair-interleaved A (16 batches)
  __shared__ float vp_lds[16 * VP_STR];      // v_posed tile
  __shared__ float res_lds[16 * RES_STR];    // results2 (16 batches x 16x12)
  __shared__ float w_lds[VPB * 16];          // skinning weights
  __shared__ float tr_lds[48];
  const int tid  = threadIdx.x;
  const int b0   = blockIdx.x * 16;
  const int vblk = blockIdx.y * VPB;

  for (int i = tid; i < 16 * KTOT; i += 256) {
    int m = i & 15, k = i >> 4;
    A_lds[((k >> 1) * 16 + m) * 2 + (k & 1)] =
        pose_ext[(size_t)(b0 + m) * KTOT + k];
  }
  for (int i = tid; i < 16 * 192; i += 256) {
    int m = i / 192, e = i - m * 192;
    res_lds[m * RES_STR + e] = res2[(size_t)(b0 + m) * 192 + e];
  }
  for (int i = tid; i < VPB * 16; i += 256) {
    int v = vblk + (i >> 4);
    w_lds[i] = (v < V_N) ? wts[v * 16 + (i & 15)] : 0.f;
  }
  if (tid < 48) tr_lds[tid] = trans_all[b0 * 3 + tid];
  __syncthreads();

  const int wv    = tid >> 5;                 // wave id: 16 vertices each
  const int lane  = tid & 31;
  const int mrow  = lane & 15;
  const int hp    = lane >> 4;                // K-pair half select
  const int mbase = hp * 8;                   // C/D tile: M = mbase + r
  const int c0w   = (vblk + wv * 16) * 3;
  v8f acc0 = {}; v8f acc1 = {}; v8f acc2 = {};

  if (c0w < BMCOLS) {                         // wave-uniform (EXEC all-ones)
#if HAVE_WMMA
    const int ccol = c0w + mrow;
    for (int kp = 0; kp < KPAIRS; kp += 2) {
      int rp = kp + hp;
      v2f a = *(const v2f*)&A_lds[(rp * 16 + mrow) * 2];
      const float* bb = bmat + ((size_t)rp * BMCOLS + ccol) * 2;
      v2f bv0 = *(const v2f*)(bb);
      v2f bv1 = *(const v2f*)(bb + 32);
      v2f bv2 = *(const v2f*)(bb + 64);
      acc0 = __builtin_amdgcn_wmma_f32_16x16x4_f32(false, a, false, bv0,
                                                   (short)0, acc0, false, false);
      acc1 = __builtin_amdgcn_wmma_f32_16x16x4_f32(false, a, false, bv1,
                                                   (short)0, acc1, false, false);
      acc2 = __builtin_amdgcn_wmma_f32_16x16x4_f32(false, a, false, bv2,
                                                   (short)0, acc2, false, false);
    }
#else
    // scalar fallback producing the same D-tile layout
    for (int k = 0; k < KTOT; ++k) {
      int rp = k >> 1, par = k & 1;
      const float* bb = bmat + ((size_t)rp * BMCOLS + c0w + mrow) * 2 + par;
      for (int r = 0; r < 8; ++r) {
        float av = A_lds[(rp * 16 + (mbase + r)) * 2 + par];
        acc0[r] += av * bb[0];
        acc1[r] += av * bb[32];
        acc2[r] += av * bb[64];
      }
    }
#endif
  }
  {
    int nb = wv * 48 + mrow;
    for (int r = 0; r < 8; ++r) {
      int m = mbase + r;                      // D layout: v[r] lanes<16 -> M=r
      vp_lds[m * VP_STR + nb + 0]  = acc0[r];
      vp_lds[m * VP_STR + nb + 16] = acc1[r];
      vp_lds[m * VP_STR + nb + 32] = acc2[r];
    }
  }
  __syncthreads();

  // ---- LBS skinning: each thread owns batch m_fix, 8 vertices
  const int m_fix = tid & 15;
  const float tx = tr_lds[m_fix*3+0], ty = tr_lds[m_fix*3+1], tz = tr_lds[m_fix*3+2];
  const float* rm = &res_lds[m_fix * RES_STR];
  const size_t bout = (size_t)(b0 + m_fix);
  for (int it = 0; it < 8; ++it) {
    int vl = (tid >> 4) + it * 16;
    int v = vblk + vl;
    if (v >= V_N) continue;
    float Tm[12];
#pragma unroll
    for (int e = 0; e < 12; ++e) Tm[e] = 0.f;
    const float* wrow = &w_lds[vl * 16];
    for (int j = 0; j < 16; ++j) {
      float wj = wrow[j];
      const float* mj = rm + j * 12;
#pragma unroll
      for (int e = 0; e < 12; ++e) Tm[e] += wj * mj[e];
    }
    float x = vp_lds[m_fix * VP_STR + vl * 3 + 0];
    float y = vp_lds[m_fix * VP_STR + vl * 3 + 1];
    float z = vp_lds[m_fix * VP_STR + vl * 3 + 2];
    float ox = (Tm[0]*x + Tm[1]*y + Tm[2]*z  + Tm[3]  + tx) * 1000.f;
    float oy = (Tm[4]*x + Tm[5]*y + Tm[6]*z  + Tm[7]  + ty) * 1000.f;
    float oz = (Tm[8]*x + Tm[9]*y + Tm[10]*z + Tm[11] + tz) * 1000.f;
    float* vo = verts_out + (bout * V_N + v) * 3;
    vo[0] = ox; vo[1] = oy; vo[2] = oz;
    int tip = -1;
    if      (v == 745) tip = 0;
    else if (v == 317) tip = 1;
    else if (v == 444) tip = 2;
    else if (v == 556) tip = 3;
    else if (v == 673) tip = 4;
    if (tip >= 0) {
      float* jo = jtr_out + (bout * 21 + (size_t)(4 * tip + 4)) * 3;
      jo[0] = ox; jo[1] = oy; jo[2] = oz;
    }
  }
}

// ----------------------------------------------------------------- launcher
extern "C" void kernel_launch(void* const* d_in, const int* in_sizes, int n_in,
                              void* d_out, int out_size, void* d_ws, size_t ws_size,
                              hipStream_t stream)
{
  const float* pc    = (const float*)d_in[0];   // th_pose_coeffs (B,48)
  const float* betas = (const float*)d_in[1];   // th_betas       (B,10)
  const float* trans = (const float*)d_in[2];   // th_trans       (B,3)
  const float* sd    = (const float*)d_in[3];   // th_shapedirs   (V,3,10)
  const float* pd    = (const float*)d_in[4];   // th_posedirs    (V,3,135)
  const float* vt    = (const float*)d_in[5];   // th_v_template  (V,3)
  const float* jr    = (const float*)d_in[6];   // th_J_regressor (16,V)
  const float* wts   = (const float*)d_in[7];   // th_weights     (V,16)
  const float* sc    = (const float*)d_in[8];   // th_selected_comps (45,45)
  const float* hm    = (const float*)d_in[9];   // th_hands_mean  (45)

  float* verts = (float*)d_out;
  float* jtr   = verts + (size_t)B_N * V_N * 3;

  float* ws   = (float*)d_ws;                   // ~7 MB total
  float* bmat = ws;                                        // 148*2352
  float* pose = bmat + (size_t)KTOT * BMCOLS;              // B*148
  float* res2 = pose + (size_t)B_N * KTOT;                 // B*192
  float* jsd  = res2 + (size_t)B_N * 192;                  // 528

  build_bmat_kernel<<<(KTOT * BMCOLS + 255) / 256, 256, 0, stream>>>(sd, pd, vt, bmat);
  build_jsd_kernel<<<(J_N * 33 + 63) / 64, 64, 0, stream>>>(jr, sd, vt, jsd);
  rig_kernel<<<B_N / 64, 64, 0, stream>>>(pc, betas, trans, sc, hm, jsd,
                                          pose, res2, jtr);
  dim3 grid(B_N / 16, (V_N + VPB - 1) / VPB);
  skin_kernel<<<grid, 256, 0, stream>>>(bmat, pose, res2, trans, wts, verts, jtr);
}